// SPR_RGCN_88648124990299
// MI455X (gfx1250) — compile-verified
//
#include <hip/hip_runtime.h>

typedef float v2f __attribute__((ext_vector_type(2)));
typedef float v8f __attribute__((ext_vector_type(8)));

// ---------------------------------------------------------------------------
// Embedding gather: h[n,:] = embed[x[n],:]   (float4 per thread)
// ---------------------------------------------------------------------------
__global__ __launch_bounds__(256) void k_embed_gather(
    const int* __restrict__ x, const float* __restrict__ embed,
    float* __restrict__ h, int N)
{
    int idx = blockIdx.x * 256 + threadIdx.x;          // over N*16 float4s
    if (idx >= N * 16) return;
    int n = idx >> 4, q = idx & 15;
    ((float4*)h)[(size_t)n * 16 + q] =
        ((const float4*)embed)[(size_t)x[n] * 16 + q];
}

// ---------------------------------------------------------------------------
// Per-(relation,dst) in-degree counts
// ---------------------------------------------------------------------------
__global__ __launch_bounds__(256) void k_count(
    const int* __restrict__ dst, const int* __restrict__ et,
    int* __restrict__ cnt, int N, int E)
{
    int e = blockIdx.x * 256 + threadIdx.x;
    if (e >= E) return;
    atomicAdd(&cnt[(size_t)et[e] * N + dst[e]], 1);
}

// Per-edge mean coefficient (same for both layers, computed once)
__global__ __launch_bounds__(256) void k_coef(
    const int* __restrict__ dst, const int* __restrict__ et,
    const int* __restrict__ cnt, float* __restrict__ coef, int N, int E)
{
    int e = blockIdx.x * 256 + threadIdx.x;
    if (e >= E) return;
    coef[e] = 1.0f / (float)cnt[(size_t)et[e] * N + dst[e]];
}

// ---------------------------------------------------------------------------
// WMMA transform: hw[r] = h @ W[r]  (r=0..2),  agg = h @ root + bias
// Block = 512 threads (16 waves), 64 nodes per block. A tile staged in LDS.
// Each wave owns exactly ONE (matrix, col-tile) combo: 32 VGPRs of B frags +
// 8 accumulators -> low register pressure, no VGPR-MSB switching.
// 16 f32 16x16x4 WMMA k-steps per 16x16 output tile, 4 row tiles per wave.
// ---------------------------------------------------------------------------
__global__ __launch_bounds__(512) void k_transform(
    const float* __restrict__ h,
    const float* __restrict__ W,     // [3][64][64]
    const float* __restrict__ root,  // [64][64]
    const float* __restrict__ bias,  // [64]
    float* __restrict__ hw,          // [3][N][64]
    float* __restrict__ agg,         // [N][64]
    int N)
{
    __shared__ float lds[64 * 66];                     // padded stride 66
    const int t = threadIdx.x;
    const int nodeBase = blockIdx.x * 64;

    // Stage 64x64 A tile into LDS (coalesced global reads, clamped tail)
    #pragma unroll
    for (int i = 0; i < 8; ++i) {
        int flat = i * 512 + t;
        int row = flat >> 6;
        int col = flat & 63;
        int gr = nodeBase + row; if (gr >= N) gr = N - 1;
        lds[row * 66 + col] = h[(size_t)gr * 64 + col];
    }
    __syncthreads();

    const int combo = t >> 5;                          // wave id == combo
    const int lane  = t & 31;
    const int l16   = lane & 15;                       // M row / N col index
    const int hi    = lane >> 4;                       // K half select

    const int r  = combo >> 2;                         // 0..2 = rel, 3 = root
    const int ct = combo & 3;                          // 16-col tile
    const float* Wsrc = (r < 3) ? (W + r * 64 * 64) : root;
    const int gcol = ct * 16 + l16;

    // Preload all 16 B fragments (K-major, lane-half selects K pair)
    v2f bfrag[16];
    #pragma unroll
    for (int kk = 0; kk < 16; ++kk) {
        int k0 = kk * 4 + 2 * hi;
        bfrag[kk].x = Wsrc[(k0 + 0) * 64 + gcol];
        bfrag[kk].y = Wsrc[(k0 + 1) * 64 + gcol];
    }

    for (int rt = 0; rt < 4; ++rt) {
        v8f acc = {};
        const float* arow = &lds[(rt * 16 + l16) * 66];
        #pragma unroll
        for (int kk = 0; kk < 16; ++kk) {
            int k0 = kk * 4 + 2 * hi;
            v2f a;
            a.x = arow[k0];
            a.y = arow[k0 + 1];
            acc = __builtin_amdgcn_wmma_f32_16x16x4_f32(
                    false, a, false, bfrag[kk], (short)0, acc,
                    false, false);
        }
        #pragma unroll
        for (int v = 0; v < 8; ++v) {
            int gRow = nodeBase + rt * 16 + v + 8 * hi;
            if (gRow < N) {
                float val = acc[v];
                if (r < 3)
                    hw[((size_t)r * N + gRow) * 64 + gcol] = val;
                else
                    agg[(size_t)gRow * 64 + gcol] = val + bias[gcol];
            }
        }
    }
}

// ---------------------------------------------------------------------------
// Edge aggregation: agg[dst] += coef[e] * hw[etype, src]
// 16 threads per edge, float4 each: 256B coalesced gather + 4 atomic f32 adds
// ---------------------------------------------------------------------------
__global__ __launch_bounds__(256) void k_edge_agg(
    const float* __restrict__ hw,
    const int* __restrict__ src, const int* __restrict__ dst,
    const int* __restrict__ et,  const float* __restrict__ coef,
    float* __restrict__ agg, int N, int E)
{
    int t = threadIdx.x;
    int e = blockIdx.x * 16 + (t >> 4);
    int c = t & 15;
    if (e >= E) return;
    int s = src[e], d = dst[e], r = et[e];
    float cf = coef[e];
    float4 v = ((const float4*)(hw + ((size_t)r * N + s) * 64))[c];
    float* out = agg + (size_t)d * 64 + (size_t)c * 4;
    atomicAdd(out + 0, v.x * cf);
    atomicAdd(out + 1, v.y * cf);
    atomicAdd(out + 2, v.z * cf);
    atomicAdd(out + 3, v.w * cf);
}

// ---------------------------------------------------------------------------
// ReLU: h = max(agg, 0)
// ---------------------------------------------------------------------------
__global__ __launch_bounds__(256) void k_relu(
    const float* __restrict__ in, float* __restrict__ out, int n4)
{
    int i = blockIdx.x * 256 + threadIdx.x;
    if (i >= n4) return;
    float4 v = ((const float4*)in)[i];
    v.x = fmaxf(v.x, 0.0f);
    v.y = fmaxf(v.y, 0.0f);
    v.z = fmaxf(v.z, 0.0f);
    v.w = fmaxf(v.w, 0.0f);
    ((float4*)out)[i] = v;
}

// ---------------------------------------------------------------------------
// Mean pool (sums + counts via atomics)
// ---------------------------------------------------------------------------
__global__ __launch_bounds__(256) void k_pool(
    const float* __restrict__ h, const int* __restrict__ batch,
    float* __restrict__ psum, float* __restrict__ pcnt, int N)
{
    int idx = blockIdx.x * 256 + threadIdx.x;          // over N*64
    if (idx >= N * 64) return;
    int n = idx >> 6, d = idx & 63;
    int g = batch[n];
    atomicAdd(&psum[(size_t)g * 64 + d], h[idx]);
    if (d == 0) atomicAdd(&pcnt[g], 1.0f);
}

// ---------------------------------------------------------------------------
// Head: out[g,c] = (psum[g,:]/max(cnt,1)) @ lin_w[:,c] + lin_b[c]
// ---------------------------------------------------------------------------
__global__ __launch_bounds__(256) void k_head(
    const float* __restrict__ psum, const float* __restrict__ pcnt,
    const float* __restrict__ lin_w, const float* __restrict__ lin_b,
    float* __restrict__ out, int G, int C)
{
    int idx = blockIdx.x * 256 + threadIdx.x;
    if (idx >= G * C) return;
    int g = idx / C, c = idx % C;
    float inv = 1.0f / fmaxf(pcnt[g], 1.0f);
    float acc = lin_b[c];
    #pragma unroll 8
    for (int j = 0; j < 64; ++j)
        acc += psum[(size_t)g * 64 + j] * inv * lin_w[j * C + c];
    out[idx] = acc;
}

// ---------------------------------------------------------------------------
extern "C" void kernel_launch(void* const* d_in, const int* in_sizes, int n_in,
                              void* d_out, int out_size, void* d_ws, size_t ws_size,
                              hipStream_t stream)
{
    const int*   x     = (const int*)d_in[0];
    const int*   eidx  = (const int*)d_in[1];   // [2,E]
    const int*   et    = (const int*)d_in[2];
    const int*   batch = (const int*)d_in[3];
    const float* embed = (const float*)d_in[4];
    const float* w1    = (const float*)d_in[5];
    const float* root1 = (const float*)d_in[6];
    const float* b1    = (const float*)d_in[7];
    const float* w2    = (const float*)d_in[8];
    const float* root2 = (const float*)d_in[9];
    const float* b2    = (const float*)d_in[10];
    const float* linw  = (const float*)d_in[11];
    const float* linb  = (const float*)d_in[12];

    const int N = in_sizes[0];
    const int E = in_sizes[2];
    const int C = in_sizes[12];
    const int G = out_size / C;
    const int* src = eidx;
    const int* dst = eidx + E;

    // Workspace layout (all f32 unless noted), ~272 MB total:
    size_t N64 = (size_t)N * 64;
    float* h    = (float*)d_ws;             // N*64
    float* hw   = h + N64;                  // 3*N*64
    float* agg  = hw + 3 * N64;             // N*64
    float* coef = agg + N64;                // E
    int*   cnt  = (int*)(coef + E);         // 3*N ints
    float* psum = (float*)(cnt + (size_t)3 * N);   // G*64
    float* pcnt = psum + (size_t)G * 64;           // G

    hipMemsetAsync(cnt, 0, (size_t)3 * N * sizeof(int), stream);
    hipMemsetAsync(psum, 0, ((size_t)G * 64 + G) * sizeof(float), stream);

    k_embed_gather<<<(N * 16 + 255) / 256, 256, 0, stream>>>(x, embed, h, N);
    k_count<<<(E + 255) / 256, 256, 0, stream>>>(dst, et, cnt, N, E);
    k_coef <<<(E + 255) / 256, 256, 0, stream>>>(dst, et, cnt, coef, N, E);

    int tblocks = (N + 63) / 64;
    int eblocks = (E + 15) / 16;
    int rblocks = (N * 16 + 255) / 256;

    // Layer 1
    k_transform<<<tblocks, 512, 0, stream>>>(h, w1, root1, b1, hw, agg, N);
    k_edge_agg <<<eblocks, 256, 0, stream>>>(hw, src, dst, et, coef, agg, N, E);
    k_relu     <<<rblocks, 256, 0, stream>>>(agg, h, N * 16);

    // Layer 2
    k_transform<<<tblocks, 512, 0, stream>>>(h, w2, root2, b2, hw, agg, N);
    k_edge_agg <<<eblocks, 256, 0, stream>>>(hw, src, dst, et, coef, agg, N, E);
    k_relu     <<<rblocks, 256, 0, stream>>>(agg, h, N * 16);

    k_pool<<<(N * 64 + 255) / 256, 256, 0, stream>>>(h, batch, psum, pcnt, N);
    k_head<<<(G * C + 255) / 256, 256, 0, stream>>>(psum, pcnt, linw, linb,
                                                    (float*)d_out, G, C);
}